// RatingConv_70892730188376
// MI455X (gfx1250) — compile-verified
//
#include <hip/hip_runtime.h>

#define NUM_USERS 100000
#define NUM_ITEMS 50000
#define NUM_EDGES 1000000
#define EMB_DIM   64

#define TPB 256
#define WAVES_PER_BLOCK (TPB / 32)
#define TILE 32

#if defined(__HIP_DEVICE_COMPILE__)
typedef __attribute__((address_space(1))) int g1_int;  // global
typedef __attribute__((address_space(3))) int l3_int;  // LDS
#endif

// ---------------------------------------------------------------------------
// Native f32 global atomic add (global_atomic_add_f32, no CAS loop).
// ---------------------------------------------------------------------------
__device__ __forceinline__ void atomic_add_f32(float* p, float v) {
#if defined(__HIP_DEVICE_COMPILE__)
    unsafeAtomicAdd(p, v);
#else
    atomicAdd(p, v);
#endif
}

// ---------------------------------------------------------------------------
// Zero a region (float4 grid-stride). All regions we zero are /4 elements.
// ---------------------------------------------------------------------------
__global__ __launch_bounds__(TPB) void zero_kernel(float4* __restrict__ p, long n4) {
    long i = (long)blockIdx.x * blockDim.x + threadIdx.x;
    const long stride = (long)gridDim.x * blockDim.x;
    const float4 z = make_float4(0.f, 0.f, 0.f, 0.f);
    for (; i < n4; i += stride) p[i] = z;
}

// ---------------------------------------------------------------------------
// Degree histogram: one thread per edge.
// ---------------------------------------------------------------------------
__global__ __launch_bounds__(TPB) void degree_kernel(const int* __restrict__ e_src,
                                                     const int* __restrict__ e_dst,
                                                     float* __restrict__ deg_u,
                                                     float* __restrict__ deg_i,
                                                     int n_edges) {
    const int i = blockIdx.x * TPB + threadIdx.x;
    if (i < n_edges) {
        atomic_add_f32(&deg_u[e_src[i]], 1.0f);
        atomic_add_f32(&deg_i[e_dst[i]], 1.0f);
    }
}

// In-place: deg -> (deg>0 ? 1/max(deg,1) : 0)   (DGL zero-degree convention)
__global__ __launch_bounds__(TPB) void invdeg_kernel(float* __restrict__ d, int n) {
    const int i = blockIdx.x * TPB + threadIdx.x;
    if (i < n) {
        const float x = d[i];
        d[i] = (x > 0.f) ? (1.0f / fmaxf(x, 1.0f)) : 0.f;
    }
}

// ---------------------------------------------------------------------------
// Edge scatter: accum[dst] += feat[src].  One wave32 per 32-edge tile.
// Indices are staged wave-locally into LDS via CDNA5 async global->LDS loads
// (ASYNCcnt path); each lane then handles 2 of the 64 embedding dims, so a
// row gather is one coalesced 256B request and the adds are native f32
// global atomics that resolve in the (fully resident) 192MB L2.
// ---------------------------------------------------------------------------
__global__ __launch_bounds__(TPB) void scatter_sum_kernel(const float* __restrict__ feat,
                                                          const int* __restrict__ e_src,
                                                          const int* __restrict__ e_dst,
                                                          float* __restrict__ accum,
                                                          int n_edges) {
    __shared__ int lds_src[TPB];
    __shared__ int lds_dst[TPB];

    const int lane  = threadIdx.x & 31;
    const int wave  = threadIdx.x >> 5;
    const int wbase = wave * 32;

    const long tile  = (long)blockIdx.x * WAVES_PER_BLOCK + wave;
    const long ebase = tile * TILE;
    if (ebase >= n_edges) return;

    const long reml = (long)n_edges - ebase;
    const int  rem  = reml > TILE ? TILE : (int)reml;
    const int  eidx = (int)ebase + lane;

#if defined(__HIP_DEVICE_COMPILE__) &&                                  \
    __has_builtin(__builtin_amdgcn_global_load_async_to_lds_b32) &&     \
    __has_builtin(__builtin_amdgcn_s_wait_asynccnt)
    if (lane < rem) {
        __builtin_amdgcn_global_load_async_to_lds_b32(
            (g1_int*)(e_src + eidx),
            (l3_int*)&lds_src[wbase + lane],
            /*offset=*/0, /*cpol=*/0);
        __builtin_amdgcn_global_load_async_to_lds_b32(
            (g1_int*)(e_dst + eidx),
            (l3_int*)&lds_dst[wbase + lane],
            /*offset=*/0, /*cpol=*/0);
    }
    __builtin_amdgcn_s_wait_asynccnt(0);
    asm volatile("" ::: "memory");  // keep LDS reads below the async wait
#else
    if (lane < rem) {
        lds_src[wbase + lane] = e_src[eidx];
        lds_dst[wbase + lane] = e_dst[eidx];
    }
#endif
    // Tile is private to this wave32 -> no __syncthreads needed.

    const int c = lane * 2;  // this lane's 2 dims of the 64-dim row
    for (int e = 0; e < rem; ++e) {
        const int s = lds_src[wbase + e];  // same-address LDS read: broadcast
        const int d = lds_dst[wbase + e];
        const float2 v = *(const float2*)(feat + ((size_t)s << 6) + c);
        float* o = accum + ((size_t)d << 6) + c;
        atomic_add_f32(o,     v.x);
        atomic_add_f32(o + 1, v.y);
    }
}

// ---------------------------------------------------------------------------
// out[n] = accum[n]*inv_deg[n] + emb[n]*sw[n]   (32 lanes x float2 per node)
// ---------------------------------------------------------------------------
__global__ __launch_bounds__(TPB) void combine_kernel(const float* __restrict__ accum,
                                                      const float* __restrict__ inv_deg,
                                                      const float* __restrict__ emb,
                                                      const float* __restrict__ sw,
                                                      float* __restrict__ out,
                                                      int n_nodes) {
    const int gid  = blockIdx.x * TPB + threadIdx.x;
    const int node = gid >> 5;
    const int lane = gid & 31;
    if (node >= n_nodes) return;

    const float inv = inv_deg[node];
    const float s   = sw[node];
    const size_t off = ((size_t)node << 6) + lane * 2;

    const float2 a = *(const float2*)(accum + off);
    const float2 x = *(const float2*)(emb + off);
    float2 r;
    r.x = fmaf(a.x, inv, x.x * s);
    r.y = fmaf(a.y, inv, x.y * s);
    *(float2*)(out + off) = r;
}

// ---------------------------------------------------------------------------
// Host-side orchestration (all on `stream`, graph-capture safe).
// ---------------------------------------------------------------------------
extern "C" void kernel_launch(void* const* d_in, const int* in_sizes, int n_in,
                              void* d_out, int out_size, void* d_ws, size_t ws_size,
                              hipStream_t stream) {
    (void)in_sizes; (void)n_in; (void)out_size; (void)ws_size;

    const float* user_emb = (const float*)d_in[0];
    const float* item_emb = (const float*)d_in[1];
    const float* u_sw     = (const float*)d_in[2];
    const float* i_sw     = (const float*)d_in[3];
    const int*   e_src    = (const int*)d_in[4];
    const int*   e_dst    = (const int*)d_in[5];

    float* out_u = (float*)d_out;                       // 100000 x 64
    float* out_i = out_u + (size_t)NUM_USERS * EMB_DIM; //  50000 x 64

    // Workspace layout (floats):
    //   [0, 6.4M)        acc_u
    //   [6.4M, 9.6M)     acc_i
    //   [9.6M, 9.7M)     deg_u -> inv_deg_u (in place)
    //   [9.7M, 9.75M)    deg_i -> inv_deg_i (in place)
    float* ws    = (float*)d_ws;
    float* acc_u = ws;
    float* acc_i = ws + (size_t)NUM_USERS * EMB_DIM;
    float* deg_u = ws + (size_t)(NUM_USERS + NUM_ITEMS) * EMB_DIM;
    float* deg_i = deg_u + NUM_USERS;

    const long acc_floats = (long)(NUM_USERS + NUM_ITEMS) * EMB_DIM;  // 9.6M
    const long deg_floats = NUM_USERS + NUM_ITEMS;                    // 150K

    const int edge_blocks    = (NUM_EDGES + TPB - 1) / TPB;                         // 3907
    const int scatter_blocks = ((NUM_EDGES + TILE - 1) / TILE + WAVES_PER_BLOCK - 1)
                               / WAVES_PER_BLOCK;                                    // 3907
    const int comb_u_blocks  = (NUM_USERS * 32 + TPB - 1) / TPB;                     // 12500
    const int comb_i_blocks  = (NUM_ITEMS * 32 + TPB - 1) / TPB;                     // 6250
    const int deg_blocks     = (int)((deg_floats + TPB - 1) / TPB);

    // ---- degrees ----
    zero_kernel<<<1024, TPB, 0, stream>>>((float4*)ws, (acc_floats + deg_floats) / 4);
    degree_kernel<<<edge_blocks, TPB, 0, stream>>>(e_src, e_dst, deg_u, deg_i, NUM_EDGES);
    invdeg_kernel<<<deg_blocks, TPB, 0, stream>>>(deg_u, (int)deg_floats);  // contiguous u then i

    // ---- round 1: u,i -> u1,i1 (stored in d_out) ----
    scatter_sum_kernel<<<scatter_blocks, TPB, 0, stream>>>(user_emb, e_src, e_dst, acc_i, NUM_EDGES);
    scatter_sum_kernel<<<scatter_blocks, TPB, 0, stream>>>(item_emb, e_dst, e_src, acc_u, NUM_EDGES);
    combine_kernel<<<comb_u_blocks, TPB, 0, stream>>>(acc_u, deg_u, user_emb, u_sw, out_u, NUM_USERS);
    combine_kernel<<<comb_i_blocks, TPB, 0, stream>>>(acc_i, deg_i, item_emb, i_sw, out_i, NUM_ITEMS);

    // ---- round 2: u1,i1 -> final (d_out in place) ----
    zero_kernel<<<1024, TPB, 0, stream>>>((float4*)ws, acc_floats / 4);
    scatter_sum_kernel<<<scatter_blocks, TPB, 0, stream>>>(out_u, e_src, e_dst, acc_i, NUM_EDGES);
    scatter_sum_kernel<<<scatter_blocks, TPB, 0, stream>>>(out_i, e_dst, e_src, acc_u, NUM_EDGES);
    combine_kernel<<<comb_u_blocks, TPB, 0, stream>>>(acc_u, deg_u, out_u, u_sw, out_u, NUM_USERS);
    combine_kernel<<<comb_i_blocks, TPB, 0, stream>>>(acc_i, deg_i, out_i, i_sw, out_i, NUM_ITEMS);
}